// HybridModel_12360915878599
// MI455X (gfx1250) — compile-verified
//
#include <hip/hip_runtime.h>
#include <math.h>

// CDNA5 / gfx1250: wave32, WMMA f32 16x16x4.
typedef float v2f __attribute__((ext_vector_type(2)));
typedef float v8f __attribute__((ext_vector_type(8)));

__device__ __forceinline__ v8f wmma_f32(v2f a, v2f b, v8f c) {
  // (neg_a, A, neg_b, B, c_mod, C, reuse_a, reuse_b)
  return __builtin_amdgcn_wmma_f32_16x16x4_f32(false, a, false, b, (short)0, c,
                                               false, false);
}

#define B_ROWS 16384
#define N_ATOM 45
#define FDIM   225   // 45 * 5

// ---------------------------------------------------------------------------
// Kernel 1: per-(b, atom) reciprocal L2 norm of the 32-feature row.
// ---------------------------------------------------------------------------
__global__ __launch_bounds__(256) void k_norm1(const float* __restrict__ x,
                                               float* __restrict__ rn1,
                                               int total) {
  int id = blockIdx.x * 256 + threadIdx.x;
  if (id >= total) return;
  const float4* p4 = (const float4*)(x + (size_t)id * 32);
  float s = 0.f;
#pragma unroll
  for (int j = 0; j < 8; ++j) {
    float4 v = p4[j];
    s += v.x * v.x + v.y * v.y + v.z * v.z + v.w * v.w;
  }
  rn1[id] = 1.0f / sqrtf(s);
}

// ---------------------------------------------------------------------------
// Kernel 2: build 45 per-atom 32x32 complex unitaries (1 layer, 5 qubits,
// CNOT range r=1). Columns of the identity evolved through the circuit in LDS.
// Output stored in "paired-k" B layout: Bp[(k>>1)*64 + 2n + (k&1)] = U[n][k].
// ---------------------------------------------------------------------------
__global__ __launch_bounds__(256) void k_buildU1(const float* __restrict__ w1,
                                                 float* __restrict__ R1p,
                                                 float* __restrict__ I1p) {
  __shared__ __align__(16) float lre[1024];  // [col*32 + i]
  __shared__ __align__(16) float lim[1024];
  const int a = blockIdx.x;
  const int tid = threadIdx.x;
  for (int idx = tid; idx < 1024; idx += 256) {
    int col = idx >> 5, i = idx & 31;
    lre[idx] = (i == col) ? 1.f : 0.f;
    lim[idx] = 0.f;
  }
  __syncthreads();
  // Rot gates (commute across distinct wires, applied sequentially anyway)
  for (int q = 0; q < 5; ++q) {
    float phi = w1[a * 15 + q * 3 + 0];
    float th  = w1[a * 15 + q * 3 + 1];
    float om  = w1[a * 15 + q * 3 + 2];
    float c, sn, ca, sa, cd, sd;
    sincosf(th * 0.5f, &sn, &c);
    sincosf((phi + om) * 0.5f, &sa, &ca);
    sincosf((phi - om) * 0.5f, &sd, &cd);
    float u00r =  ca * c,  u00i = -sa * c;
    float u01r = -cd * sn, u01i = -sd * sn;
    float u10r =  cd * sn, u10i = -sd * sn;
    float u11r =  ca * c,  u11i =  sa * c;
    const int mk = 1 << (4 - q);
    for (int t = tid; t < 512; t += 256) {
      int col = t & 31, p = t >> 5;
      int i0 = ((p & ~(mk - 1)) << 1) | (p & (mk - 1));
      int i1 = i0 | mk;
      int o0 = col * 32 + i0, o1 = col * 32 + i1;
      float a0r = lre[o0], a0i = lim[o0], a1r = lre[o1], a1i = lim[o1];
      lre[o0] = u00r * a0r - u00i * a0i + u01r * a1r - u01i * a1i;
      lim[o0] = u00r * a0i + u00i * a0r + u01r * a1i + u01i * a1r;
      lre[o1] = u10r * a0r - u10i * a0i + u11r * a1r - u11i * a1i;
      lim[o1] = u10r * a0i + u10i * a0r + u11r * a1i + u11i * a1r;
    }
    __syncthreads();
  }
  // CNOT(w, (w+1)%5), sequential (they do not all commute)
  for (int w = 0; w < 5; ++w) {
    int tgt = (w + 1) % 5;
    int cb = 4 - w;
    int mk = 1 << (4 - tgt);
    for (int t = tid; t < 512; t += 256) {
      int col = t & 31, p = t >> 5;
      int i0 = ((p & ~(mk - 1)) << 1) | (p & (mk - 1));
      if ((i0 >> cb) & 1) {
        int i1 = i0 | mk;
        int o0 = col * 32 + i0, o1 = col * 32 + i1;
        float tr = lre[o0]; lre[o0] = lre[o1]; lre[o1] = tr;
        float ti = lim[o0]; lim[o0] = lim[o1]; lim[o1] = ti;
      }
    }
    __syncthreads();
  }
  // Write paired-k layout: column j of LDS holds U[:, j]; B[k][n] = U[n][k]
  for (int idx = tid; idx < 1024; idx += 256) {
    int j = idx >> 5, i = idx & 31;
    int dst = a * 1024 + (j >> 1) * 64 + (i << 1) + (j & 1);
    R1p[dst] = lre[idx];
    I1p[dst] = lim[idx];
  }
}

// ---------------------------------------------------------------------------
// Kernel 3: build the 512x512 complex unitary of the 9-qubit, 3-layer circuit
// (CNOT ranges r = 1,2,3). 8 identity columns per workgroup, 32 KB LDS.
// ---------------------------------------------------------------------------
__global__ __launch_bounds__(256) void k_buildU2(const float* __restrict__ w2,
                                                 float* __restrict__ R2p,
                                                 float* __restrict__ I2p) {
  __shared__ __align__(16) float lre[8 * 512];  // [lc*512 + i]
  __shared__ __align__(16) float lim[8 * 512];
  const int base = blockIdx.x * 8;
  const int tid = threadIdx.x;
  for (int idx = tid; idx < 4096; idx += 256) {
    int lc = idx >> 9, i = idx & 511;
    lre[idx] = (i == base + lc) ? 1.f : 0.f;
    lim[idx] = 0.f;
  }
  __syncthreads();
  for (int l = 0; l < 3; ++l) {
    for (int q = 0; q < 9; ++q) {
      float phi = w2[l * 27 + q * 3 + 0];
      float th  = w2[l * 27 + q * 3 + 1];
      float om  = w2[l * 27 + q * 3 + 2];
      float c, sn, ca, sa, cd, sd;
      sincosf(th * 0.5f, &sn, &c);
      sincosf((phi + om) * 0.5f, &sa, &ca);
      sincosf((phi - om) * 0.5f, &sd, &cd);
      float u00r =  ca * c,  u00i = -sa * c;
      float u01r = -cd * sn, u01i = -sd * sn;
      float u10r =  cd * sn, u10i = -sd * sn;
      float u11r =  ca * c,  u11i =  sa * c;
      const int mk = 1 << (8 - q);
      for (int t = tid; t < 2048; t += 256) {
        int lc = t >> 8, p = t & 255;
        int i0 = ((p & ~(mk - 1)) << 1) | (p & (mk - 1));
        int i1 = i0 | mk;
        int o0 = lc * 512 + i0, o1 = lc * 512 + i1;
        float a0r = lre[o0], a0i = lim[o0], a1r = lre[o1], a1i = lim[o1];
        lre[o0] = u00r * a0r - u00i * a0i + u01r * a1r - u01i * a1i;
        lim[o0] = u00r * a0i + u00i * a0r + u01r * a1i + u01i * a1r;
        lre[o1] = u10r * a0r - u10i * a0i + u11r * a1r - u11i * a1i;
        lim[o1] = u10r * a0i + u10i * a0r + u11r * a1i + u11i * a1r;
      }
      __syncthreads();
    }
    const int r = l + 1;  // l % (9-1) + 1
    for (int w = 0; w < 9; ++w) {
      int tgt = (w + r) % 9;
      int cb = 8 - w;
      int mk = 1 << (8 - tgt);
      for (int t = tid; t < 2048; t += 256) {
        int lc = t >> 8, p = t & 255;
        int i0 = ((p & ~(mk - 1)) << 1) | (p & (mk - 1));
        if ((i0 >> cb) & 1) {
          int i1 = i0 | mk;
          int o0 = lc * 512 + i0, o1 = lc * 512 + i1;
          float tr = lre[o0]; lre[o0] = lre[o1]; lre[o1] = tr;
          float ti = lim[o0]; lim[o0] = lim[o1]; lim[o1] = ti;
        }
      }
      __syncthreads();
    }
  }
  for (int idx = tid; idx < 4096; idx += 256) {
    int lc = idx >> 9, i = idx & 511;
    int j = base + lc;
    size_t dst = (size_t)(j >> 1) * 1024 + (i << 1) + (j & 1);
    R2p[dst] = lre[idx];
    I2p[dst] = lim[idx];
  }
}

// ---------------------------------------------------------------------------
// Kernel 4: stage-1 GEMMs + expectations. One wave per (atom, 16-row tile):
//   Ar = Psi @ ReU^T, Ai = Psi @ ImU^T (K=32, two 16-col N-tiles), via
//   v_wmma_f32_16x16x4_f32; probs = Ar^2+Ai^2; 5 signed reductions -> feat.
// ---------------------------------------------------------------------------
__global__ __launch_bounds__(256) void k_gemm1(const float* __restrict__ x,
                                               const float* __restrict__ rn1,
                                               const float* __restrict__ R1p,
                                               const float* __restrict__ I1p,
                                               float* __restrict__ feat) {
  const int lane = threadIdx.x & 31;
  const int wid  = threadIdx.x >> 5;
  const int a  = blockIdx.y;
  const int b0 = (blockIdx.x * 8 + wid) * 16;
  const int nlo = lane & 15;           // N column within tile; also A-frag row
  const int khalf = (lane >> 4) << 1;  // upper half-wave owns k+2,k+3
  const float rn = rn1[(b0 + nlo) * N_ATOM + a];
  const float* xrow = x + ((size_t)(b0 + nlo) * N_ATOM + a) * 32;
  const float* Bre = R1p + a * 1024;
  const float* Bim = I1p + a * 1024;

  v8f cr0 = {0.f,0.f,0.f,0.f,0.f,0.f,0.f,0.f};
  v8f cr1 = cr0, ci0 = cr0, ci1 = cr0;
#pragma unroll
  for (int k0 = 0; k0 < 32; k0 += 4) {
    const int kb = k0 + khalf;
    v2f af;
    af.x = xrow[kb] * rn;
    af.y = xrow[kb + 1] * rn;
    const float* bre = Bre + (kb >> 1) * 64;
    const float* bim = Bim + (kb >> 1) * 64;
    v2f br0 = *(const v2f*)(bre + (nlo << 1));
    v2f br1 = *(const v2f*)(bre + ((16 + nlo) << 1));
    v2f bi0 = *(const v2f*)(bim + (nlo << 1));
    v2f bi1 = *(const v2f*)(bim + ((16 + nlo) << 1));
    cr0 = wmma_f32(af, br0, cr0);
    cr1 = wmma_f32(af, br1, cr1);
    ci0 = wmma_f32(af, bi0, ci0);
    ci1 = wmma_f32(af, bi1, ci1);
  }
  // Expectations: exps[m][w] = sum_i probs[m][i] * (1 - 2*bit_{4-w}(i))
  const int i0 = nlo, i1 = 16 + nlo;
  const int mrow = ((lane >> 4) << 3);  // +0 or +8
#pragma unroll
  for (int v = 0; v < 8; ++v) {
    float p0 = cr0[v] * cr0[v] + ci0[v] * ci0[v];
    float p1 = cr1[v] * cr1[v] + ci1[v] * ci1[v];
#pragma unroll
    for (int w = 0; w < 5; ++w) {
      float s0 = ((i0 >> (4 - w)) & 1) ? -1.f : 1.f;
      float s1 = ((i1 >> (4 - w)) & 1) ? -1.f : 1.f;
      float val = p0 * s0 + p1 * s1;
      val += __shfl_xor(val, 1, 32);   // butterfly within each half-wave
      val += __shfl_xor(val, 2, 32);
      val += __shfl_xor(val, 4, 32);
      val += __shfl_xor(val, 8, 32);
      if (nlo == 0)
        feat[(size_t)(b0 + v + mrow) * FDIM + a * 5 + w] = val;
    }
  }
}

// ---------------------------------------------------------------------------
// Kernel 5: stage-2 input reciprocal norm (225 features + 287 pads of 0.001).
// ---------------------------------------------------------------------------
__global__ __launch_bounds__(256) void k_norm2(const float* __restrict__ feat,
                                               float* __restrict__ rn2) {
  const int lane = threadIdx.x & 31;
  const int b = blockIdx.x * 8 + (threadIdx.x >> 5);
  const float* row = feat + (size_t)b * FDIM;
  float s = 0.f;
  for (int j = lane; j < FDIM; j += 32) { float v = row[j]; s += v * v; }
  s += __shfl_xor(s, 1, 32);
  s += __shfl_xor(s, 2, 32);
  s += __shfl_xor(s, 4, 32);
  s += __shfl_xor(s, 8, 32);
  s += __shfl_xor(s, 16, 32);
  if (lane == 0) rn2[b] = 1.0f / sqrtf(s + 287.0e-6f);  // 287 * 0.001^2
}

// ---------------------------------------------------------------------------
// Kernel 6: stage-2 GEMM (16 x 512 x 512, Re+Im) + fused head.
// 8 waves share a 16-row Psi tile in LDS; each wave owns 4 N-tiles.
// out[b] = sigmoid( sum_i probs[b][i] * s(i) + bias ),
//   s(i) = sum_w W[w] * (1 - 2*bit_{8-w}(i)).
// ---------------------------------------------------------------------------
__global__ __launch_bounds__(256) void k_gemm2(const float* __restrict__ feat,
                                               const float* __restrict__ rn2,
                                               const float* __restrict__ R2p,
                                               const float* __restrict__ I2p,
                                               const float* __restrict__ W,
                                               const float* __restrict__ bias,
                                               float* __restrict__ out) {
  __shared__ __align__(16) float psA[16 * 516];  // stride 516: no bank conflicts
  __shared__ float partial[16];
  __shared__ float Ws[9];
  const int tid = threadIdx.x;
  const int lane = tid & 31, wid = tid >> 5;
  const int b0 = blockIdx.x * 16;
  if (tid < 9) Ws[tid] = W[tid];
  if (tid < 16) partial[tid] = 0.f;
  for (int idx = tid; idx < 16 * 512; idx += 256) {
    int m = idx >> 9, j = idx & 511;
    float v = (j < FDIM) ? feat[(size_t)(b0 + m) * FDIM + j] : 0.001f;
    psA[m * 516 + j] = v * rn2[b0 + m];
  }
  __syncthreads();

  const int nlo = lane & 15;
  const int khalf = (lane >> 4) << 1;
  const int mrow = ((lane >> 4) << 3);
  for (int nt = wid * 4; nt < wid * 4 + 4; ++nt) {
    const int ncol = nt * 16 + nlo;
    float s = 0.f;
#pragma unroll
    for (int w = 0; w < 9; ++w)
      s += Ws[w] * (((ncol >> (8 - w)) & 1) ? -1.f : 1.f);
    v8f cr = {0.f,0.f,0.f,0.f,0.f,0.f,0.f,0.f};
    v8f ci = cr;
    for (int k0 = 0; k0 < 512; k0 += 4) {
      const int kb = k0 + khalf;
      v2f af = *(const v2f*)(psA + nlo * 516 + kb);
      v2f br = *(const v2f*)(R2p + (size_t)(kb >> 1) * 1024 + (ncol << 1));
      v2f bi = *(const v2f*)(I2p + (size_t)(kb >> 1) * 1024 + (ncol << 1));
      cr = wmma_f32(af, br, cr);
      ci = wmma_f32(af, bi, ci);
    }
#pragma unroll
    for (int v = 0; v < 8; ++v) {
      float val = (cr[v] * cr[v] + ci[v] * ci[v]) * s;
      val += __shfl_xor(val, 1, 32);
      val += __shfl_xor(val, 2, 32);
      val += __shfl_xor(val, 4, 32);
      val += __shfl_xor(val, 8, 32);
      if (nlo == 0) atomicAdd(&partial[v + mrow], val);
    }
  }
  __syncthreads();
  if (tid < 16) {
    float z = partial[tid] + bias[0];
    out[b0 + tid] = 1.0f / (1.0f + expf(-z));
  }
}

// ---------------------------------------------------------------------------
extern "C" void kernel_launch(void* const* d_in, const int* in_sizes, int n_in,
                              void* d_out, int out_size, void* d_ws,
                              size_t ws_size, hipStream_t stream) {
  (void)in_sizes; (void)n_in; (void)out_size; (void)ws_size;
  const float* x    = (const float*)d_in[0];  // (16384, 45, 32)
  const float* w1   = (const float*)d_in[1];  // (45, 1, 5, 3)
  const float* w2   = (const float*)d_in[2];  // (3, 9, 3)
  const float* W    = (const float*)d_in[3];  // (1, 9)
  const float* bias = (const float*)d_in[4];  // (1,)
  float* out = (float*)d_out;                 // (16384, 1)

  float* ws  = (float*)d_ws;                  // ~20.2 MB of f32 scratch
  float* rn1 = ws;                                 // 16384*45
  float* R1p = rn1 + (size_t)B_ROWS * N_ATOM;      // 45*1024
  float* I1p = R1p + 45 * 1024;                    // 45*1024
  float* R2p = I1p + 45 * 1024;                    // 512*512
  float* I2p = R2p + 512 * 512;                    // 512*512
  float* ft  = I2p + 512 * 512;                    // 16384*225
  float* rn2 = ft + (size_t)B_ROWS * FDIM;         // 16384

  k_norm1<<<(B_ROWS * N_ATOM + 255) / 256, 256, 0, stream>>>(x, rn1,
                                                             B_ROWS * N_ATOM);
  k_buildU1<<<N_ATOM, 256, 0, stream>>>(w1, R1p, I1p);
  k_buildU2<<<64, 256, 0, stream>>>(w2, R2p, I2p);
  k_gemm1<<<dim3(B_ROWS / 128, N_ATOM), 256, 0, stream>>>(x, rn1, R1p, I1p, ft);
  k_norm2<<<B_ROWS / 8, 256, 0, stream>>>(ft, rn2);
  k_gemm2<<<B_ROWS / 16, 256, 0, stream>>>(ft, rn2, R2p, I2p, W, bias, out);
}